// KeypointAlignmentLossL2_37855841747427
// MI455X (gfx1250) — compile-verified
//
#include <hip/hip_runtime.h>
#include <hip/hip_bf16.h>

// KeypointAlignmentLossL2 on gfx1250 (MI455X).
// Memory-bound bilinear gather. Per-keypoint dot(f1,f2) computed as the
// diagonal of a 16x16 cross-product via V_WMMA_F32_16X16X4_F32 (exact f32,
// accumulator lives in D registers across the loop); |f1|^2 and |f2|^2
// accumulate per-lane with packed FMAs (each lane holds both operands for
// its own keypoint/channels). One 256-thread block (8 wave32) per 16
// keypoints; each wave reduces a 96-channel slice (constant 24-iteration
// loop -> scalar control); partials fold through LDS; wave 0 runs the
// normalize/L2/mask epilogue. 4096 waves, one atomic pair per block.

typedef __attribute__((ext_vector_type(2))) float v2f;
typedef __attribute__((ext_vector_type(8))) float v8f;
// 4-byte-aligned float2 so clang may emit global_load_b64 at dword alignment
typedef float f2u __attribute__((ext_vector_type(2), aligned(4)));

#define EPSF 1e-12f
#define Bsz 8
#define Csz 768
#define Hsz 64
#define Wsz 64
#define Nsz 1024
#define PLANE (Hsz * Wsz) /* 4096 */
#define SPLITS 8           /* waves per block, channel slices */
#define CPW (Csz / SPLITS) /* 96 channels per wave */
#define ITERS (CPW / 4)    /* 24 chunks of K=4, same for every wave */

// Row-pair tap description. Each bilinear sample is evaluated as
//   wa*r0.x + wb*r0.y + wc*r1.x + wd*r1.y
// where r0 = float2 at (rowY0, xb), r1 = float2 at (rowY1, xb).
// When x0 == W-1 the base is shifted left by one and the weights are
// remapped, so the fetch never crosses the tensor end and OOB taps get
// weight zero (matches grid_sample zeros padding).
struct Taps {
    int o0, o1;               // row base offsets within one channel plane
    float wa, wb, wc, wd;     // weights for r0.x, r0.y, r1.x, r1.y
};

__device__ __forceinline__ Taps make_taps(float x, float y) {
    const float x0f = floorf(x), y0f = floorf(y);
    const float wx = x - x0f, wy = y - y0f;
    const int x0 = (int)x0f, y0 = (int)y0f;
    const bool sx = (x0 >= Wsz - 1);   // x1 out of bounds
    const bool sy = (y0 >= Hsz - 1);   // y1 out of bounds
    const int xb  = x0 - (sx ? 1 : 0);
    const int y1c = sy ? y0 : (y0 + 1);

    const float w00 = (1.f - wx) * (1.f - wy);
    const float w01 = sx ? 0.f : wx * (1.f - wy);
    const float w10 = sy ? 0.f : (1.f - wx) * wy;
    const float w11 = (sx || sy) ? 0.f : wx * wy;

    Taps t;
    t.o0 = y0 * Wsz + xb;
    t.o1 = y1c * Wsz + xb;
    // shift-fold: when sx, r.y holds the x0 element and r.x gets weight 0
    t.wa = sx ? 0.f : w00;
    t.wb = sx ? w00 : w01;
    t.wc = sx ? 0.f : w10;
    t.wd = sx ? w10 : w11;
    return t;
}

__device__ __forceinline__ float pick8(const v8f& v, int i) {
    switch (i & 7) {
        case 0: return v[0]; case 1: return v[1];
        case 2: return v[2]; case 3: return v[3];
        case 4: return v[4]; case 5: return v[5];
        case 6: return v[6]; default: return v[7];
    }
}

__device__ __forceinline__ f2u ld2(const float* __restrict__ p) {
    return *(const f2u*)p;
}

__global__ __launch_bounds__(256) void kal_main_kernel(
    const float* __restrict__ feat1, const float* __restrict__ feat2,
    const float* __restrict__ kp1, const float* __restrict__ kp2,
    const unsigned char* __restrict__ mask1, const unsigned char* __restrict__ mask2,
    float* __restrict__ sums)
{
    __shared__ float part[SPLITS][16][3]; // [wave][keypoint][dot,n1,n2]

    const int lane = threadIdx.x & 31;
    const int wave = threadIdx.x >> 5;   // 0..7 = channel slice
    const int kbase = blockIdx.x * 16;   // 16 keypoints per block
    const int m    = lane & 15;          // my keypoint in group
    const int half = lane >> 4;          // K-half per WMMA f32 layout
    const int kglob = kbase + m;
    const int b = kglob >> 10;           // Nsz == 1024
    const int n = kglob & (Nsz - 1);

    const float* kp1p = kp1 + (size_t)(b * Nsz + n) * 2;
    const float* kp2p = kp2 + (size_t)(b * Nsz + n) * 2;
    const Taps t1 = make_taps(kp1p[0], kp1p[1]);
    const Taps t2 = make_taps(kp2p[0], kp2p[1]);

    // Slice + K-half folded into four tap base pointers; the compiler
    // strength-reduces the per-iteration bumps into immediate offsets.
    const size_t cbase = (size_t)(wave * CPW + 2 * half) * PLANE;
    const float* pa0 = feat1 + (size_t)b * Csz * PLANE + cbase + t1.o0;
    const float* pa1 = feat1 + (size_t)b * Csz * PLANE + cbase + t1.o1;
    const float* pb0 = feat2 + (size_t)b * Csz * PLANE + cbase + t2.o0;
    const float* pb1 = feat2 + (size_t)b * Csz * PLANE + cbase + t2.o1;

    v8f accD = {};                 // WMMA acc; diag -> partial dot(f1_i, f2_i)
    v2f n1v = {0.f, 0.f};          // per-lane partial |f1|^2 (two channels)
    v2f n2v = {0.f, 0.f};          // per-lane partial |f2|^2

    for (int it = 0; it < ITERS; ++it) { // constant trip count, scalar control
        // ---- issue all 8 row loads first (independent dests -> clause) ----
        const f2u a0r0 = ld2(pa0);
        const f2u a0r1 = ld2(pa1);
        const f2u a1r0 = ld2(pa0 + PLANE);
        const f2u a1r1 = ld2(pa1 + PLANE);
        const f2u b0r0 = ld2(pb0);
        const f2u b0r1 = ld2(pb1);
        const f2u b1r0 = ld2(pb0 + PLANE);
        const f2u b1r1 = ld2(pb1 + PLANE);

        if (it + 1 < ITERS) { // scalar-uniform guard: stays in bounds
            __builtin_prefetch(pa0 + 4 * (size_t)PLANE, 0, 1);
            __builtin_prefetch(pb0 + 4 * (size_t)PLANE, 0, 1);
        }

        // ---- then all the math ----
        v2f a, bv;
        a.x  = fmaf(a0r0.x, t1.wa, fmaf(a0r0.y, t1.wb,
               fmaf(a0r1.x, t1.wc, a0r1.y * t1.wd)));
        a.y  = fmaf(a1r0.x, t1.wa, fmaf(a1r0.y, t1.wb,
               fmaf(a1r1.x, t1.wc, a1r1.y * t1.wd)));
        bv.x = fmaf(b0r0.x, t2.wa, fmaf(b0r0.y, t2.wb,
               fmaf(b0r1.x, t2.wc, b0r1.y * t2.wd)));
        bv.y = fmaf(b1r0.x, t2.wa, fmaf(b1r0.y, t2.wb,
               fmaf(b1r1.x, t2.wc, b1r1.y * t2.wd)));

        // Cross-product dot via WMMA (diagonal used); norms per-lane (pk_fma).
        accD = __builtin_amdgcn_wmma_f32_16x16x4_f32(false, a, false, bv, (short)0, accD, false, false);
        n1v.x = fmaf(a.x,  a.x,  n1v.x);
        n1v.y = fmaf(a.y,  a.y,  n1v.y);
        n2v.x = fmaf(bv.x, bv.x, n2v.x);
        n2v.y = fmaf(bv.y, bv.y, n2v.y);

        pa0 += 4 * (size_t)PLANE;
        pa1 += 4 * (size_t)PLANE;
        pb0 += 4 * (size_t)PLANE;
        pb1 += 4 * (size_t)PLANE;
    }

    // Norm partials: fold channel pair in-lane, then fold the two K-halves
    // (lane m and lane m+16 hold disjoint channel sets of keypoint m).
    const float n1h = n1v.x + n1v.y;
    const float n2h = n2v.x + n2v.y;
    const float n1s = n1h + __shfl_xor(n1h, 16, 32);
    const float n2s = n2h + __shfl_xor(n2h, 16, 32);
    if (lane < 16) {
        part[wave][lane][1] = n1s;
        part[wave][lane][2] = n2s;
    }

    // Diagonal (i,i) of a 16x16 f32 C/D tile: i<8 -> VGPR i, lane i;
    // i>=8 -> VGPR i-8, lane i+16. So lanes 0..7 and 24..31 hold diagonals.
    int di = -1, dreg = 0;
    if (lane < 8)        { di = lane;      dreg = lane; }
    else if (lane >= 24) { di = lane - 16; dreg = lane - 24; }
    if (di >= 0) {
        part[wave][di][0] = pick8(accD, dreg);
    }
    __syncthreads();

    // Wave 0: fold the 8 channel-slice partials per keypoint, run epilogue.
    if (wave == 0) {
        float c_l2 = 0.f, c_valid = 0.f;
        if (lane < 16) {
            float dv = 0.f, n1 = 0.f, n2 = 0.f;
            #pragma unroll
            for (int s = 0; s < SPLITS; ++s) {
                dv += part[s][lane][0];
                n1 += part[s][lane][1];
                n2 += part[s][lane][2];
            }
            const int kg = kbase + lane;
            const int bb = kg >> 10;
            const int nn = kg & (Nsz - 1);
            const float valid =
                (mask1[bb * Nsz + nn] && mask2[bb * Nsz + nn]) ? 1.f : 0.f;
            const float m1 = fmaxf(sqrtf(n1), EPSF);
            const float m2 = fmaxf(sqrtf(n2), EPSF);
            // ||f1/m1 - f2/m2||^2 = n1/m1^2 + n2/m2^2 - 2 dot/(m1 m2)
            const float l2 = n1 / (m1 * m1) + n2 / (m2 * m2) - 2.f * dv / (m1 * m2);
            c_l2 = l2 * valid;
            c_valid = valid;
        }
        // wave32 reduction (lanes >=16 contribute zeros), one atomic pair/block
        for (int off = 16; off > 0; off >>= 1) {
            c_l2    += __shfl_xor(c_l2, off, 32);
            c_valid += __shfl_xor(c_valid, off, 32);
        }
        if (lane == 0) {
            atomicAdd(&sums[0], c_l2);
            atomicAdd(&sums[1], c_valid);
        }
    }
}

__global__ void kal_init_kernel(float* __restrict__ sums) {
    if (threadIdx.x < 2) sums[threadIdx.x] = 0.f;
}

__global__ void kal_final_kernel(const float* __restrict__ sums, float* __restrict__ out) {
    if (threadIdx.x == 0) {
        const float sv = sums[1];
        out[0] = (sv == 0.f) ? 0.f : sums[0] / fmaxf(sv, 1.f);
    }
}

extern "C" void kernel_launch(void* const* d_in, const int* in_sizes, int n_in,
                              void* d_out, int out_size, void* d_ws, size_t ws_size,
                              hipStream_t stream) {
    const float* feat1 = (const float*)d_in[0];
    const float* feat2 = (const float*)d_in[1];
    const float* kp1   = (const float*)d_in[2];
    const float* kp2   = (const float*)d_in[3];
    // jax bool arrays are 1 byte/element
    const unsigned char* m1 = (const unsigned char*)d_in[4];
    const unsigned char* m2 = (const unsigned char*)d_in[5];
    float* sums = (float*)d_ws;   // [0]=sum_l2, [1]=sum_valid
    float* out  = (float*)d_out;

    (void)in_sizes; (void)n_in; (void)out_size; (void)ws_size;

    kal_init_kernel<<<1, 32, 0, stream>>>(sums);

    const int total_kp = Bsz * Nsz;     // 8192
    const int grid     = total_kp / 16; // 512 blocks, 8 waves each -> 4096 waves
    kal_main_kernel<<<grid, 256, 0, stream>>>(feat1, feat2, kp1, kp2, m1, m2, sums);

    kal_final_kernel<<<1, 32, 0, stream>>>(sums, out);
}